// IsingTransformerLayer_87514253623890
// MI455X (gfx1250) — compile-verified
//
#include <hip/hip_runtime.h>
#include <math.h>

// ---------------------------------------------------------------------------
// Ising transformer layer magnetizations for MI455X (gfx1250, wave32, WMMA).
// Pipeline (all GEMMs via V_WMMA_F32_16X16X4_F32):
//   xn = per-head normalize(x)
//   QK = xn @ w_qk^T                                (NT GEMM, vector-staged)
//   sim_h = Q_h @ K_h^T -> J = softmax/sqrt(D)      (NT GEMM + row softmax)
//   V = diag(t) - J ; X = V^{-1} via Newton-Schulz  (NN GEMMs, TDM-staged)
//   U = X@h, W = X^T@h                              (NN-TDM / TN GEMMs)
//   G_J from X, W U^T, diag terms                   (NT GEMM + elementwise)
//   softmax backward (rowwise), g_q/g_k GEMMs, head-block projection,
//   assemble output (direct h-term folded in).
// NN GEMMs use the Tensor Data Mover: 2D tile descriptors with hardware LDS
// row padding, double-buffered LDS, TENSORcnt-tracked overlap of DMA+WMMA.
// ---------------------------------------------------------------------------

#define N_TOK 1024
#define NH    12
#define DH    64
#define DIM   768
#define DIM2  1536
static const long N2 = (long)N_TOK * N_TOK;

typedef __attribute__((ext_vector_type(2))) float v2f;
typedef __attribute__((ext_vector_type(8))) float v8f;
typedef __attribute__((ext_vector_type(4))) float f4;
typedef __attribute__((ext_vector_type(4))) unsigned int u32x4;
typedef __attribute__((ext_vector_type(8))) int i32x8;
typedef __attribute__((ext_vector_type(4))) int i32x4;

// ---------------------------------------------------------------------------
// TDM: issue a 2D tile load (height rows x width f32, row stride in floats)
// into LDS at byte offset ldsOff, with LDS row padding of padAmount DWORDs
// after every padInterval-coded DWORD interval (interval code: 3 -> 16 DW,
// 5 -> 64 DW; amount code: 3 -> 4 DW).  D# packing per CDNA5 ISA ch.8.
// This toolchain exposes the 6-arg builtin (clang-23 / therock-10.0 flavor):
//   (u32x4 g0, i32x8 g1, i32x4 g2, i32x4 g3, i32x8 g4, i32 cpol)
// Groups 2-4 are zero for a plain 2D tile.
// ---------------------------------------------------------------------------
__device__ __forceinline__ void tdm_load_2d(unsigned ldsOff, const float* gptr,
                                            unsigned width, unsigned height,
                                            unsigned strideF, unsigned padIntCode,
                                            unsigned padAmtCode) {
  unsigned long long ga = (unsigned long long)(uintptr_t)gptr;
  u32x4 g0;
  g0[0] = 1u;                                    // count=1, user mode
  g0[1] = ldsOff;                                // lds_addr [63:32]
  g0[2] = (unsigned)ga;                          // global_addr lo
  g0[3] = (unsigned)(ga >> 32) | (2u << 30);     // global_addr hi | type=2
  i32x8 g1;
  g1[0] = (int)((2u << 16)                       // data_size = 4B
                | (1u << 20)                     // pad_enable
                | (padIntCode << 22) | (padAmtCode << 25));
  g1[1] = (int)(width << 16);                    // tensor_dim0 lo16 @ [63:48]
  g1[2] = (int)(height << 16);                   // tdim0 hi=0; tensor_dim1 lo16
  g1[3] = (int)(width << 16);                    // tdim1 hi=0; tile_dim0
  g1[4] = (int)height;                           // tile_dim1; tile_dim2=0
  g1[5] = (int)strideF;                          // tensor_dim0_stride lo32
  g1[6] = 0;                                     // stride hi; tdim1_stride lo
  g1[7] = 0;
  i32x4 z4 = {0, 0, 0, 0};
  i32x8 z8 = {0, 0, 0, 0, 0, 0, 0, 0};
  __builtin_amdgcn_tensor_load_to_lds(g0, g1, z4, z4, z8, 0);
}

// ---------------------------------------------------------------------------
// NN GEMM, TDM-staged + double-buffered: C = alpha*A[MxK]*B[KxN] + beta*C
// Grid (N/64, M/64, batch), 256 threads = 8 waves, wave tile 16x32.
// ---------------------------------------------------------------------------
__global__ __launch_bounds__(256) void gemm_wmma_f32_tdm(
    const float* __restrict__ Ab, const float* __restrict__ Bb,
    float* __restrict__ Cb, int K, int lda, int ldb, int ldc,
    long sA, long sB, long sC, float alpha, float beta) {
  __shared__ float As[2][64][20];   // 16 data + 4 pad DWORDs per row
  __shared__ float Bs[2][16][68];   // 64 data + 4 pad DWORDs per row

  const float* A = Ab + (long)blockIdx.z * sA;
  const float* B = Bb + (long)blockIdx.z * sB;
  float* C = Cb + (long)blockIdx.z * sC;

  const int row0 = blockIdx.y * 64;
  const int col0 = blockIdx.x * 64;
  const int t = threadIdx.x;
  const int wave = t >> 5, lane = t & 31;
  const int wm = wave >> 1, wn = wave & 1;
  const int hl = lane >> 4, lm = lane & 15;

  const unsigned asOff[2] = {(unsigned)(uintptr_t)&As[0][0][0],
                             (unsigned)(uintptr_t)&As[1][0][0]};
  const unsigned bsOff[2] = {(unsigned)(uintptr_t)&Bs[0][0][0],
                             (unsigned)(uintptr_t)&Bs[1][0][0]};

  v8f acc0 = {};
  v8f acc1 = {};

  const int nt = K >> 4;
  if (t == 0) {   // prologue: tile 0 in flight (TDM ignores EXEC; one wave issues)
    tdm_load_2d(asOff[0], A + (long)row0 * lda, 16u, 64u, (unsigned)lda, 3u, 3u);
    tdm_load_2d(bsOff[0], B + col0, 64u, 16u, (unsigned)ldb, 5u, 3u);
  }
  for (int kt = 0; kt < nt; ++kt) {
    if (kt + 1 < nt) {
      if (t == 0) {
        const int k0n = (kt + 1) << 4;
        const int pn = (kt + 1) & 1;
        tdm_load_2d(asOff[pn], A + (long)row0 * lda + k0n, 16u, 64u,
                    (unsigned)lda, 3u, 3u);
        tdm_load_2d(bsOff[pn], B + (long)k0n * ldb + col0, 64u, 16u,
                    (unsigned)ldb, 5u, 3u);
      }
      __builtin_amdgcn_s_wait_tensorcnt(2);   // tile kt done; kt+1 may run
    } else {
      __builtin_amdgcn_s_wait_tensorcnt(0);
    }
    __syncthreads();                          // publish LDS to all waves

    const int ph = kt & 1;
    const int arow = wm * 16 + lm;
    const int bcol = wn * 32 + lm;
#pragma unroll
    for (int kk = 0; kk < 16; kk += 4) {
      v2f a;
      a.x = As[ph][arow][kk + 2 * hl];
      a.y = As[ph][arow][kk + 2 * hl + 1];
      v2f b0, b1;
      b0.x = Bs[ph][kk + 2 * hl][bcol];
      b0.y = Bs[ph][kk + 2 * hl + 1][bcol];
      b1.x = Bs[ph][kk + 2 * hl][bcol + 16];
      b1.y = Bs[ph][kk + 2 * hl + 1][bcol + 16];
      acc0 = __builtin_amdgcn_wmma_f32_16x16x4_f32(false, a, false, b0,
                                                   (short)0, acc0, false, false);
      acc1 = __builtin_amdgcn_wmma_f32_16x16x4_f32(false, a, false, b1,
                                                   (short)0, acc1, false, false);
    }
    __syncthreads();                          // safe to overwrite buf[ph] next
  }

  const int crow0 = row0 + wm * 16 + hl * 8;
  const int ccol = col0 + wn * 32 + lm;
#pragma unroll
  for (int j = 0; j < 8; ++j) {
    long idx = (long)(crow0 + j) * ldc + ccol;
    float v0 = alpha * acc0[j];
    float v1 = alpha * acc1[j];
    if (beta != 0.0f) {
      v0 += beta * C[idx];
      v1 += beta * C[idx + 16];
    }
    C[idx] = v0;
    C[idx + 16] = v1;
  }
}

// ---------------------------------------------------------------------------
// Vector-staged WMMA GEMM for transposed flavors (TDM cannot transpose).
// OPA==1: logical A[m][k] = Astore[k*lda + m]; OPB==1: B[k][n] = Bstore[n*ldb+k]
// ---------------------------------------------------------------------------
template <int OPA, int OPB>
__global__ __launch_bounds__(256) void gemm_wmma_f32(
    const float* __restrict__ Ab, const float* __restrict__ Bb,
    float* __restrict__ Cb, int K, int lda, int ldb, int ldc,
    long sA, long sB, long sC, float alpha, float beta) {
  __shared__ float As[64][20];
  __shared__ float Bs[16][68];

  const float* A = Ab + (long)blockIdx.z * sA;
  const float* B = Bb + (long)blockIdx.z * sB;
  float* C = Cb + (long)blockIdx.z * sC;

  const int row0 = blockIdx.y * 64;
  const int col0 = blockIdx.x * 64;
  const int t = threadIdx.x;
  const int wave = t >> 5, lane = t & 31;
  const int wm = wave >> 1, wn = wave & 1;
  const int hl = lane >> 4, lm = lane & 15;

  v8f acc0 = {};
  v8f acc1 = {};

  for (int k0 = 0; k0 < K; k0 += 16) {
    if (OPA == 0) {
      int m = t >> 2, kq = (t & 3) << 2;
      f4 v = *(const f4*)(A + (long)(row0 + m) * lda + (k0 + kq));
      *(f4*)&As[m][kq] = v;
    } else {
      int k = t >> 4, mq = (t & 15) << 2;
      f4 v = *(const f4*)(A + (long)(k0 + k) * lda + (row0 + mq));
      As[mq + 0][k] = v.x; As[mq + 1][k] = v.y;
      As[mq + 2][k] = v.z; As[mq + 3][k] = v.w;
    }
    if (OPB == 0) {
      int k = t >> 4, nq = (t & 15) << 2;
      f4 v = *(const f4*)(B + (long)(k0 + k) * ldb + (col0 + nq));
      *(f4*)&Bs[k][nq] = v;
    } else {
      int n = t >> 2, kq = (t & 3) << 2;
      f4 v = *(const f4*)(B + (long)(col0 + n) * ldb + (k0 + kq));
      Bs[kq + 0][n] = v.x; Bs[kq + 1][n] = v.y;
      Bs[kq + 2][n] = v.z; Bs[kq + 3][n] = v.w;
    }
    if (k0 + 16 < K) {
      __builtin_prefetch(A + (long)(row0 + (t >> 2)) * lda + (k0 + 16), 0, 1);
      __builtin_prefetch(B + (long)(k0 + 16 + (t >> 4)) * ldb + col0, 0, 1);
    }
    __syncthreads();

    const int arow = wm * 16 + lm;
    const int bcol = wn * 32 + lm;
#pragma unroll
    for (int kk = 0; kk < 16; kk += 4) {
      v2f a;
      a.x = As[arow][kk + 2 * hl];
      a.y = As[arow][kk + 2 * hl + 1];
      v2f b0, b1;
      b0.x = Bs[kk + 2 * hl][bcol];
      b0.y = Bs[kk + 2 * hl + 1][bcol];
      b1.x = Bs[kk + 2 * hl][bcol + 16];
      b1.y = Bs[kk + 2 * hl + 1][bcol + 16];
      acc0 = __builtin_amdgcn_wmma_f32_16x16x4_f32(false, a, false, b0,
                                                   (short)0, acc0, false, false);
      acc1 = __builtin_amdgcn_wmma_f32_16x16x4_f32(false, a, false, b1,
                                                   (short)0, acc1, false, false);
    }
    __syncthreads();
  }

  const int crow0 = row0 + wm * 16 + hl * 8;
  const int ccol = col0 + wn * 32 + lm;
#pragma unroll
  for (int j = 0; j < 8; ++j) {
    long idx = (long)(crow0 + j) * ldc + ccol;
    float v0 = alpha * acc0[j];
    float v1 = alpha * acc1[j];
    if (beta != 0.0f) {
      v0 += beta * C[idx];
      v1 += beta * C[idx + 16];
    }
    C[idx] = v0;
    C[idx + 16] = v1;
  }
}

// ---------------------------------------------------------------------------
// Elementwise / reduction helper kernels
// ---------------------------------------------------------------------------
__global__ void k_normalize(const float* __restrict__ x, float* __restrict__ xn) {
  int blk = blockIdx.x;              // N_TOK * NH blocks
  int i = blk / NH, h = blk % NH;
  int d = threadIdx.x;               // 64 threads
  __shared__ float red[64];
  float v = x[(long)i * DIM + h * DH + d];
  red[d] = v * v;
  __syncthreads();
  for (int s = 32; s > 0; s >>= 1) {
    if (d < s) red[d] += red[d + s];
    __syncthreads();
  }
  xn[(long)i * DIM + h * DH + d] = v * rsqrtf(red[0]);
}

__global__ __launch_bounds__(256) void k_softmax(float* __restrict__ J) {
  long row = blockIdx.x;             // NH*N rows
  float* p = J + row * N_TOK;
  int t = threadIdx.x;
  __shared__ float red[256];
  float vals[4];
  float vmax = -1e30f;
#pragma unroll
  for (int q = 0; q < 4; ++q) {
    vals[q] = p[t + q * 256];
    vmax = fmaxf(vmax, vals[q]);
  }
  red[t] = vmax;
  __syncthreads();
  for (int s = 128; s > 0; s >>= 1) {
    if (t < s) red[t] = fmaxf(red[t], red[t + s]);
    __syncthreads();
  }
  vmax = red[0];
  __syncthreads();
  float sum = 0.f;
#pragma unroll
  for (int q = 0; q < 4; ++q) {
    vals[q] = __expf(vals[q] - vmax);
    sum += vals[q];
  }
  red[t] = sum;
  __syncthreads();
  for (int s = 128; s > 0; s >>= 1) {
    if (t < s) red[t] += red[t + s];
    __syncthreads();
  }
  float inv = 1.0f / (red[0] * 8.0f);   // softmax / sqrt(D), sqrt(64)=8
#pragma unroll
  for (int q = 0; q < 4; ++q) p[t + q * 256] = vals[q] * inv;
}

__global__ void k_buildV(const float* __restrict__ J, float* __restrict__ V) {
  long idx = (long)blockIdx.x * 256 + threadIdx.x;   // NH*N2 exact
  long h = idx / N2;
  long rem = idx - h * N2;
  long r = rem / N_TOK, c = rem - r * N_TOK;
  const float* Jh = J + h * N2;
  float v = -Jh[r * N_TOK + c];
  if (r == c) {
    float jii = Jh[r * N_TOK + r];
    float ti = 0.5f * (0.5f + sqrtf(1.25f + 2.0f * jii));
    v += ti;
  }
  V[idx] = v;
}

__global__ void k_setX0(const float* __restrict__ J, float* __restrict__ X) {
  long idx = (long)blockIdx.x * 256 + threadIdx.x;
  long h = idx / N2;
  long rem = idx - h * N2;
  long r = rem / N_TOK, c = rem - r * N_TOK;
  float v = 0.f;
  if (r == c) {
    float jii = J[h * N2 + r * N_TOK + r];
    float ti = 0.5f * (0.5f + sqrtf(1.25f + 2.0f * jii));
    v = 1.0f / ti;
  }
  X[idx] = v;
}

__global__ void k_E(float* __restrict__ T) {   // T <- 2I - T (in place)
  long idx = (long)blockIdx.x * 256 + threadIdx.x;
  long rem = idx % N2;
  long r = rem / N_TOK, c = rem - r * N_TOK;
  float v = -T[idx];
  if (r == c) v += 2.0f;
  T[idx] = v;
}

__global__ void k_diag(const float* __restrict__ J, const float* __restrict__ X,
                       const float* __restrict__ T, float* __restrict__ Bv,
                       float* __restrict__ Sv) {
  int idx = blockIdx.x * 256 + threadIdx.x;   // NH*N
  int h = idx / N_TOK, i = idx - h * N_TOK;
  long base = (long)h * N2 + (long)i * N_TOK + i;
  Sv[idx] = sqrtf(1.25f + 2.0f * J[base]);
  Bv[idx] = -1.0f + 0.5f * X[base] - T[base];   // T = -1/4 W U^T
}

__global__ void k_gbuild(const float* __restrict__ T, const float* __restrict__ X,
                         const float* __restrict__ Bv, const float* __restrict__ Sv,
                         float* __restrict__ G) {
  long idx = (long)blockIdx.x * 256 + threadIdx.x;
  long h = idx / N2;
  long rem = idx - h * N2;
  long r = rem / N_TOK, c = rem - r * N_TOK;
  float v = T[idx] - 0.5f * X[h * N2 + c * N_TOK + r];   // -1/2 V^{-T}
  if (r == c) {
    int ri = (int)(h * N_TOK + r);
    v += 0.5f * Bv[ri] / Sv[ri];
  }
  G[idx] = v;
}

__global__ __launch_bounds__(256) void k_smbwd(const float* __restrict__ J,
                                               float* __restrict__ G) {
  long row = blockIdx.x;             // NH*N rows
  const float* jp = J + row * N_TOK;
  float* gp = G + row * N_TOK;
  int t = threadIdx.x;
  __shared__ float red[256];
  float jv[4], gv[4];
  float acc = 0.f;
#pragma unroll
  for (int q = 0; q < 4; ++q) {
    jv[q] = jp[t + q * 256];
    gv[q] = gp[t + q * 256];
    acc += jv[q] * gv[q];
  }
  red[t] = acc;
  __syncthreads();
  for (int s = 128; s > 0; s >>= 1) {
    if (t < s) red[t] += red[t + s];
    __syncthreads();
  }
  float r = red[0];
#pragma unroll
  for (int q = 0; q < 4; ++q)
    gp[t + q * 256] = jv[q] * gv[q] - 8.0f * jv[q] * r;   // sqrt(D)=8
}

__global__ void k_assemble(const float* __restrict__ GH, const float* __restrict__ U,
                           const float* __restrict__ W, const float* __restrict__ Bv,
                           const float* __restrict__ Sv, const float* __restrict__ XN,
                           float* __restrict__ out) {
  int idx = blockIdx.x * 256 + threadIdx.x;   // NH*N*DH
  int h = idx / (N_TOK * DH);
  int rem = idx - h * N_TOK * DH;
  int i = rem / DH, d = rem - i * DH;
  float u = XN[(long)i * DIM + h * DH + d];
  int ri = h * N_TOK + i;
  float v = GH[idx] - 0.25f * (U[idx] + W[idx]) + 0.5f * Bv[ri] * u / Sv[ri];
  out[(long)i * DIM + h * DH + d] = v;
}

// ---------------------------------------------------------------------------
// Host launcher
// ---------------------------------------------------------------------------
extern "C" void kernel_launch(void* const* d_in, const int* in_sizes, int n_in,
                              void* d_out, int out_size, void* d_ws, size_t ws_size,
                              hipStream_t stream) {
  const float* x = (const float*)d_in[0];     // [1024, 768]
  const float* wqk = (const float*)d_in[1];   // [1536, 768]
  float* out = (float*)d_out;                 // [1024, 768]
  float* ws = (float*)d_ws;

  float* XN = ws;                    // 786432
  float* QK = XN + 786432;           // 1572864
  float* Jb = QK + 1572864;          // 12582912
  float* Vb = Jb + 12582912;         // 12582912 (later reused as G)
  float* X1 = Vb + 12582912;         // 12582912
  float* X2 = X1 + 12582912;         // 12582912
  float* Tb = X2 + 12582912;         // 12582912
  float* Ub = Tb + 12582912;         // 786432
  float* Wb = Ub + 786432;           // 786432
  float* GQ = Wb + 786432;           // 786432
  float* GK = GQ + 786432;           // 786432
  float* GH = GK + 786432;           // 786432
  float* Bv = GH + 786432;           // 12288
  float* Sv = Bv + 12288;            // 12288

  const long HD64 = (long)N_TOK * DH;         // 65536
  const long WSUB = 64L * DIM + 64;           // 49216: head-block stride in w_qk

  // 1) per-head unit-normalize tokens
  k_normalize<<<N_TOK * NH, 64, 0, stream>>>(x, XN);

  // 2) QK = XN @ w_qk^T  (NT): [1024x768]x[1536x768]^T -> [1024x1536]
  gemm_wmma_f32<0, 1><<<dim3(DIM2 / 64, N_TOK / 64, 1), 256, 0, stream>>>(
      XN, wqk, QK, DIM, DIM, DIM, DIM2, 0, 0, 0, 1.0f, 0.0f);

  // 3) sim_h = Q_h @ K_h^T (NT, batched over heads) -> Jb, then softmax/sqrt(D)
  gemm_wmma_f32<0, 1><<<dim3(16, 16, NH), 256, 0, stream>>>(
      QK, QK + DIM, Jb, DH, DIM2, DIM2, N_TOK, 64, 64, N2, 1.0f, 0.0f);
  k_softmax<<<NH * N_TOK, 256, 0, stream>>>(Jb);

  // 4) V = diag(t) - J ; X0 = diag(1/t)
  k_buildV<<<49152, 256, 0, stream>>>(Jb, Vb);
  k_setX0<<<49152, 256, 0, stream>>>(Jb, X1);

  // 5) Newton-Schulz: X <- X (2I - V X), 5 iterations (||R0||~0.15, quadratic)
  float* Xc = X1;
  float* Xn = X2;
  for (int it = 0; it < 5; ++it) {
    gemm_wmma_f32_tdm<<<dim3(16, 16, NH), 256, 0, stream>>>(
        Vb, Xc, Tb, N_TOK, N_TOK, N_TOK, N_TOK, N2, N2, N2, 1.0f, 0.0f);
    k_E<<<49152, 256, 0, stream>>>(Tb);
    gemm_wmma_f32_tdm<<<dim3(16, 16, NH), 256, 0, stream>>>(
        Xc, Tb, Xn, N_TOK, N_TOK, N_TOK, N_TOK, N2, N2, N2, 1.0f, 0.0f);
    float* tmp = Xc; Xc = Xn; Xn = tmp;
  }

  // 6) U = X @ h (NN, TDM), W = X^T @ h (TN, vector-staged)
  gemm_wmma_f32_tdm<<<dim3(1, 16, NH), 256, 0, stream>>>(
      Xc, XN, Ub, N_TOK, N_TOK, DIM, DH, N2, 64, HD64, 1.0f, 0.0f);
  gemm_wmma_f32<1, 0><<<dim3(1, 16, NH), 256, 0, stream>>>(
      Xc, XN, Wb, N_TOK, N_TOK, DIM, DH, N2, 64, HD64, 1.0f, 0.0f);

  // 7) Tb = -1/4 W @ U^T (NT, K=64); diag terms b,s; G_J assembled into Vb
  gemm_wmma_f32<0, 1><<<dim3(16, 16, NH), 256, 0, stream>>>(
      Wb, Ub, Tb, DH, DH, DH, N_TOK, HD64, HD64, N2, -0.25f, 0.0f);
  k_diag<<<48, 256, 0, stream>>>(Jb, Xc, Tb, Bv, Sv);
  k_gbuild<<<49152, 256, 0, stream>>>(Tb, Xc, Bv, Sv, Vb);

  // 8) softmax backward (in place on G = Vb)
  k_smbwd<<<NH * N_TOK, 256, 0, stream>>>(Jb, Vb);

  // 9) g_q = g_sim @ K_h (NN, TDM) ; g_k = g_sim^T @ Q_h (TN, vector-staged)
  gemm_wmma_f32_tdm<<<dim3(1, 16, NH), 256, 0, stream>>>(
      Vb, QK + DIM, GQ, N_TOK, N_TOK, DIM2, DH, N2, 64, HD64, 1.0f, 0.0f);
  gemm_wmma_f32<1, 0><<<dim3(1, 16, NH), 256, 0, stream>>>(
      Vb, QK, GK, N_TOK, N_TOK, DIM2, DH, N2, 64, HD64, 1.0f, 0.0f);

  // 10) GH = g_q @ Wq_sub + g_k @ Wk_sub  (64x64 head blocks of w_qk, NN TDM)
  gemm_wmma_f32_tdm<<<dim3(1, 16, NH), 256, 0, stream>>>(
      GQ, wqk, GH, DH, DH, DIM, DH, HD64, WSUB, HD64, 1.0f, 0.0f);
  gemm_wmma_f32_tdm<<<dim3(1, 16, NH), 256, 0, stream>>>(
      GK, wqk + (long)DIM * DIM, GH, DH, DH, DIM, DH, HD64, WSUB, HD64, 1.0f, 1.0f);

  // 11) out = GH - 1/4 (U+W) + 1/2 (b/s) h   (head-interleaved layout)
  k_assemble<<<3072, 256, 0, stream>>>(GH, Ub, Wb, Bv, Sv, XN, out);
}